// FullAffinity_59133109732236
// MI455X (gfx1250) — compile-verified
//
#include <hip/hip_runtime.h>
#include <math.h>

// Problem constants (fixed by setup_inputs)
#define HW   4096   // spatial positions (both tar and ref)
#define CCH  128    // channels
#define NCTX 8      // reference frames
#define TT   16     // target rows per workgroup (one WMMA M tile)
#define NWAVE 16    // waves per workgroup
#define BLK  (NWAVE * 32)

// B staging: sub-stage = 256 columns x 64 channels, row stride padded to 264
#define BCOLS   256
#define BROWS   64
#define BPITCH  264                      // 256 + 8 DWORD pad (bank-conflict-free)
#define BUF_FLT (BROWS * BPITCH)         // 16896 floats = 67584 B per buffer

typedef __attribute__((ext_vector_type(2))) float f32x2;
typedef __attribute__((ext_vector_type(8))) float f32x8;
typedef __attribute__((ext_vector_type(4))) unsigned int u32x4;
typedef __attribute__((ext_vector_type(8))) int i32x8;
typedef __attribute__((ext_vector_type(4))) int i32x4;

// ---------------------------------------------------------------------------
// Kernel 1: inverse L2 norms.
// ---------------------------------------------------------------------------
__global__ __launch_bounds__(256) void norms_kernel(const float* __restrict__ ft,
                                                    const float* __restrict__ fr,
                                                    float* __restrict__ tinv,
                                                    float* __restrict__ rinv) {
  int tid = blockIdx.x * blockDim.x + threadIdx.x;
  if (tid < HW) {
    const float4* p = (const float4*)(ft + (size_t)tid * CCH);
    float s = 0.f;
#pragma unroll
    for (int i = 0; i < CCH / 4; ++i) {
      float4 v = p[i];
      s += v.x * v.x + v.y * v.y + v.z * v.z + v.w * v.w;
    }
    tinv[tid] = 1.0f / fmaxf(sqrtf(s), 1e-12f);
  } else if (tid < HW + NCTX * HW) {
    int idx = tid - HW;
    int n = idx >> 12;
    int r = idx & (HW - 1);
    const float* p = fr + (size_t)n * CCH * HW + r;
    float s = 0.f;
#pragma unroll 8
    for (int c = 0; c < CCH; ++c) {
      float v = p[(size_t)c * HW];
      s += v * v;
    }
    rinv[idx] = 1.0f / fmaxf(sqrtf(s), 1e-12f);
  }
}

// ---------------------------------------------------------------------------
// TDM: DMA one 256x64 f32 tile (row stride HW) from global into LDS, with an
// 8-DWORD pad after every 256-DWORD row (LDS pitch = 264 floats).
// D# layout per CDNA5 ISA.
// ---------------------------------------------------------------------------
__device__ __forceinline__ void tdm_load_tile(const float* gptr, unsigned lds_off) {
  unsigned long long ga = (unsigned long long)(uintptr_t)(const void*)gptr;
  u32x4 g0;
  g0[0] = 1u;                                      // count=1 (valid), no gather
  g0[1] = lds_off;                                 // lds_addr (bytes)
  g0[2] = (unsigned)ga;                            // global_addr[31:0]
  g0[3] = (unsigned)((ga >> 32) & 0x01FFFFFFull)   // global_addr[56:32]
          | (2u << 30);                            // type = 2 ("image")
  i32x8 g1;
  g1[0] = (int)((2u << 16)      // data_size = 4 bytes
                | (1u << 20)    // pad_enable
                | (7u << 22)    // pad_interval: every 256 DWORDs (one row)
                | (7u << 25));  // pad_amount: 8 DWORDs
  g1[1] = (int)((HW & 0xFFFFu) << 16);              // tensor_dim0 [15:0]
  g1[2] = (int)((HW >> 16) | ((CCH & 0xFFFFu) << 16)); // dim0 hi / dim1 lo
  g1[3] = (int)((CCH >> 16) | ((unsigned)BCOLS << 16)); // dim1 hi / tile_dim0
  g1[4] = (int)BROWS;                               // tile_dim1=64, tile_dim2=0
  g1[5] = (int)HW;                                  // tensor_dim0_stride lo32
  g1[6] = 0;                                        // stride hi / dim1_stride lo
  g1[7] = 0;
  i32x4 z4;
  for (int e = 0; e < 4; ++e) z4[e] = 0;
#if defined(__clang_major__) && __clang_major__ >= 23
  i32x8 z8;
  for (int e = 0; e < 8; ++e) z8[e] = 0;
  __builtin_amdgcn_tensor_load_to_lds(g0, g1, z4, z4, z8, 0);
#else
  __builtin_amdgcn_tensor_load_to_lds(g0, g1, z4, z4, 0);
#endif
}

// ---------------------------------------------------------------------------
// Kernel 2: fused (normalized GEMM) + softmax over r, one output write.
// Grid: (HW/TT, NCTX). Block: 512 threads = 16 waves (wave32).
// Wave w, stage i owns the 16x16 tile at columns 256*i + 16*w, rows t0..t0+15.
// B tiles are TDM-DMA'd into double-buffered LDS, overlapped with WMMA.
// Channel half h is the (rolled) outer loop -> only 16 A-fragments live;
// the stage loop i is fully unrolled so acc[] register-promotes.
// ---------------------------------------------------------------------------
__global__ __launch_bounds__(BLK) void affinity_kernel(const float* __restrict__ ft,
                                                       const float* __restrict__ fr,
                                                       const float* __restrict__ tinv,
                                                       const float* __restrict__ rinv,
                                                       float* __restrict__ out) {
  // Manual LDS layout (floats):
  //   [0, 16896)        B buffer 0          (reused as output staging later)
  //   [16896, 33792)    B buffer 1
  //   [33792, 35840)    A tile (16 x 128)
  //   [35840, 36096)    reduction scratch (16 rows x 16 waves)
  //   [36096, 36112)    per-row stat (max / sum)
  //   [36112, 36128)    target inverse norms
  __shared__ __align__(16) float smf[2 * BUF_FLT + TT * CCH + 16 * 16 + 16 + 16];
  float* lds_a     = smf + 2 * BUF_FLT;
  float* lds_red   = lds_a + TT * CCH;
  float* lds_stat  = lds_red + 16 * 16;
  float* lds_ti    = lds_stat + 16;
  float* lds_stage = smf;                       // aliases B buf 0 (dead by then)

  const int tid  = threadIdx.x;
  const int wave = tid >> 5;
  const int lane = tid & 31;
  const int hi   = lane >> 4;
  const int l16  = lane & 15;
  const int koff = hi * 2;
  const int t0   = blockIdx.x * TT;
  const int n    = blockIdx.y;
  const float* frn = fr + (size_t)n * CCH * HW;
  const unsigned lds_base = (unsigned)(uintptr_t)(void*)smf;  // LDS byte offset

  // Prologue: kick off TDM for sub-stage 0 (stage 0, channels 0..63) -> buf 0.
  if (wave == 0) {
    tdm_load_tile(frn, lds_base);
  }

  // Load A tile (16 x 128 floats) cooperatively: 4 floats per thread.
  {
    int row = tid >> 5;
    int col = (tid & 31) * 4;
    *(float4*)&lds_a[row * CCH + col] =
        *(const float4*)&ft[(size_t)(t0 + row) * CCH + col];
  }
  if (tid < TT) lds_ti[tid] = tinv[t0 + tid];
  __syncthreads();

  f32x8 acc[16];
#pragma unroll
  for (int i = 0; i < 16; ++i)
#pragma unroll
    for (int e = 0; e < 8; ++e) acc[i][e] = 0.0f;

  // Main loop: channel half h (rolled outer) x stage i (fully unrolled).
#pragma unroll 1
  for (int h = 0; h < 2; ++h) {
    // A fragments for this channel half only (16 x f32x2 = 32 VGPRs).
    f32x2 afrag[16];
#pragma unroll
    for (int k = 0; k < 16; ++k)
      afrag[k] = *(const f32x2*)&lds_a[l16 * CCH + h * BROWS + 4 * k + koff];

#pragma unroll
    for (int i = 0; i < 16; ++i) {
      const int s = h * 16 + i;
      if (wave == 0) {
        __builtin_amdgcn_s_wait_tensorcnt(0);   // sub-stage s landed in LDS
      }
      // Single barrier per sub-stage: publishes buffer s to all waves AND
      // guarantees buffer (s+1)&1 (read in sub-stage s-1, all ds reads
      // waited before their WMMAs) is no longer in use before the prefetch
      // below overwrites it.
      __syncthreads();
      if (wave == 0 && s + 1 < 32) {
        const int s2 = s + 1;
        const int h2 = s2 >> 4;
        const int i2 = s2 & 15;
        tdm_load_tile(frn + (size_t)h2 * BROWS * HW + (size_t)i2 * BCOLS,
                      lds_base + (s2 & 1) * (BUF_FLT * 4));
      }
      // WMMA over 16 K-chunks of this sub-stage.
      const float* bb = smf + (s & 1) * BUF_FLT + wave * 16 + l16;
#pragma unroll
      for (int k = 0; k < 16; ++k) {
        const int c0 = 4 * k + koff;             // channel row within buffer
        f32x2 b;
        b.x = bb[c0 * BPITCH];
        b.y = bb[(c0 + 1) * BPITCH];
        acc[i] = __builtin_amdgcn_wmma_f32_16x16x4_f32(
            false, afrag[k], false, b, (short)0, acc[i], false, false);
      }
    }
  }

  // --- fold in normalizations: score *= tinv[row] * rinv[col] ---
  float ti[8];
#pragma unroll
  for (int v = 0; v < 8; ++v) ti[v] = lds_ti[v + 8 * hi];

  float mx[8];
#pragma unroll
  for (int v = 0; v < 8; ++v) mx[v] = -INFINITY;
#pragma unroll
  for (int i = 0; i < 16; ++i) {
    float rv = rinv[(size_t)n * HW + i * 256 + wave * 16 + l16];
#pragma unroll
    for (int v = 0; v < 8; ++v) {
      acc[i][v] *= ti[v] * rv;
      mx[v] = fmaxf(mx[v], acc[i][v]);
    }
  }

  // --- row max: reduce across the 16 lanes of each half-wave ---
#pragma unroll
  for (int sh = 8; sh >= 1; sh >>= 1)
#pragma unroll
    for (int v = 0; v < 8; ++v)
      mx[v] = fmaxf(mx[v], __shfl_xor(mx[v], sh, 32));
  if (l16 == 0)
#pragma unroll
    for (int v = 0; v < 8; ++v) lds_red[(v + 8 * hi) * 16 + wave] = mx[v];
  __syncthreads();
  if (tid < 16) {
    float m = lds_red[tid * 16];
    for (int w = 1; w < 16; ++w) m = fmaxf(m, lds_red[tid * 16 + w]);
    lds_stat[tid] = m;
  }
  __syncthreads();
  float rowmax[8];
#pragma unroll
  for (int v = 0; v < 8; ++v) rowmax[v] = lds_stat[v + 8 * hi];

  // --- exp + row sum ---
  float sm[8];
#pragma unroll
  for (int v = 0; v < 8; ++v) sm[v] = 0.0f;
#pragma unroll
  for (int i = 0; i < 16; ++i)
#pragma unroll
    for (int v = 0; v < 8; ++v) {
      float e = __expf(acc[i][v] - rowmax[v]);
      acc[i][v] = e;
      sm[v] += e;
    }
#pragma unroll
  for (int sh = 8; sh >= 1; sh >>= 1)
#pragma unroll
    for (int v = 0; v < 8; ++v) sm[v] += __shfl_xor(sm[v], sh, 32);
  if (l16 == 0)
#pragma unroll
    for (int v = 0; v < 8; ++v) lds_red[(v + 8 * hi) * 16 + wave] = sm[v];
  __syncthreads();
  if (tid < 16) {
    float ssum = 0.f;
    for (int w = 0; w < 16; ++w) ssum += lds_red[tid * 16 + w];
    lds_stat[tid] = ssum;
  }
  __syncthreads();
  float rs[8];
#pragma unroll
  for (int v = 0; v < 8; ++v) rs[v] = 1.0f / lds_stat[v + 8 * hi];

  // --- normalize + coalesced store via LDS staging (aliases dead B buf 0) ---
  // Fully unrolled: acc[i] must keep constant indices to stay in VGPRs.
  const int orow = tid >> 5;
  const int ocol = (tid & 31) * 8;
#pragma unroll
  for (int i = 0; i < 16; ++i) {
#pragma unroll
    for (int v = 0; v < 8; ++v)
      lds_stage[(v + 8 * hi) * 256 + wave * 16 + l16] = acc[i][v] * rs[v];
    __syncthreads();
    size_t g = ((size_t)n * HW + t0 + orow) * HW + (size_t)i * 256 + ocol;
    float4 p0 = *(float4*)&lds_stage[orow * 256 + ocol];
    float4 p1 = *(float4*)&lds_stage[orow * 256 + ocol + 4];
    *(float4*)&out[g]     = p0;
    *(float4*)&out[g + 4] = p1;
    __syncthreads();
  }
}

// ---------------------------------------------------------------------------
extern "C" void kernel_launch(void* const* d_in, const int* in_sizes, int n_in,
                              void* d_out, int out_size, void* d_ws, size_t ws_size,
                              hipStream_t stream) {
  const float* ft = (const float*)d_in[0];   // [4096, 128]
  const float* fr = (const float*)d_in[1];   // [8, 128, 4096]
  float* out  = (float*)d_out;               // [8, 4096, 4096]
  float* tinv = (float*)d_ws;                // 4096 floats
  float* rinv = tinv + HW;                   // 8*4096 floats

  int norm_threads = HW + NCTX * HW;         // 36864
  norms_kernel<<<(norm_threads + 255) / 256, 256, 0, stream>>>(ft, fr, tinv, rinv);
  affinity_kernel<<<dim3(HW / TT, NCTX), BLK, 0, stream>>>(ft, fr, tinv, rinv, out);
}